// GCN_Encoder_16853451670135
// MI455X (gfx1250) — compile-verified
//
#include <hip/hip_runtime.h>

typedef float v2f __attribute__((ext_vector_type(2)));
typedef float v8f __attribute__((ext_vector_type(8)));
typedef int   v4i __attribute__((vector_size(16)));   // matches builtin's V4i param

#define NNODES 50000
#define NEDGES 800000
#define FDIM   128
#define XS_STRIDE 132   // 16-row X panel, padded (4*lm mod 64 distinct banks)
#define WS_STRIDE 132   // W rows, padded, float4-aligned rows

#define GLOBAL_AS __attribute__((address_space(1)))
#define LDS_AS    __attribute__((address_space(3)))

// gfx1250 async global->LDS path (ASYNCcnt); falls back to synchronous staging
// if the builtins aren't declared by this toolchain.
#if defined(__has_builtin)
#if __has_builtin(__builtin_amdgcn_global_load_async_to_lds_b128) && \
    __has_builtin(__builtin_amdgcn_s_wait_asynccnt)
#define HAVE_ASYNC_LDS 1
#endif
#endif
#ifndef HAVE_ASYNC_LDS
#define HAVE_ASYNC_LDS 0
#endif

// ---------------- degree / normalization ----------------

__global__ void deg_init_kernel(float* __restrict__ deg, int n) {
    int i = blockIdx.x * blockDim.x + threadIdx.x;
    if (i < n) deg[i] = 1.0f;   // self-loop contributes 1
}

__global__ void deg_edge_kernel(const long long* __restrict__ dst,
                                float* __restrict__ deg, int e) {
    int i = blockIdx.x * blockDim.x + threadIdx.x;
    if (i < e) unsafeAtomicAdd(&deg[(int)dst[i]], 1.0f);
}

__global__ void rsqrt_kernel(float* __restrict__ deg, int n) {
    int i = blockIdx.x * blockDim.x + threadIdx.x;
    if (i < n) deg[i] = rsqrtf(deg[i]);   // deg >= 1 always
}

// ---------------- GEMM: H[N x 128] = X[N x 128] @ W[128 x 128] ----------------
// One block = 8 waves = one 16-row panel; wave w computes the 16x16 tile at
// columns [16w, 16w+16). Full W staged in LDS in ONE shot (76 KB static LDS —
// CDNA5 allows up to 320 KB per workgroup; 4 blocks/WGP occupancy).

__global__ __launch_bounds__(256) void gemm_wmma_kernel(
        const float* __restrict__ X, const float* __restrict__ W,
        float* __restrict__ H) {
    __shared__ float Ws[128 * WS_STRIDE];  // 67584 B
    __shared__ float Xs[16 * XS_STRIDE];   //  8448 B

    const int tid = threadIdx.x;
    const int m0  = blockIdx.x * 16;

#if HAVE_ASYNC_LDS
    // DMA global->LDS with no VGPR round-trip; tracked by ASYNCcnt.
    for (int idx = tid; idx < (128 * FDIM) / 4; idx += 256) {
        int g = idx * 4;
        int r = g >> 7, c = g & 127;
        __builtin_amdgcn_global_load_async_to_lds_b128(
            (GLOBAL_AS v4i*)const_cast<float*>(W + g),
            (LDS_AS v4i*)(Ws + r * WS_STRIDE + c), 0, 0);
    }
    for (int idx = tid; idx < (16 * FDIM) / 4; idx += 256) {
        int g = idx * 4;
        int r = g >> 7, c = g & 127;
        __builtin_amdgcn_global_load_async_to_lds_b128(
            (GLOBAL_AS v4i*)const_cast<float*>(X + (size_t)(m0 + r) * FDIM + c),
            (LDS_AS v4i*)(Xs + r * XS_STRIDE + c), 0, 0);
    }
    __builtin_amdgcn_s_wait_asynccnt(0);
    __syncthreads();
#else
    for (int idx = tid; idx < (128 * FDIM) / 4; idx += 256) {
        int g = idx * 4;
        int r = g >> 7, c = g & 127;
        float4 v = *(const float4*)(W + g);
        *(float4*)(Ws + r * WS_STRIDE + c) = v;
    }
    for (int idx = tid; idx < (16 * FDIM) / 4; idx += 256) {
        int g = idx * 4;
        int r = g >> 7, c = g & 127;
        float4 v = *(const float4*)(X + (size_t)(m0 + r) * FDIM + c);
        *(float4*)(Xs + r * XS_STRIDE + c) = v;
    }
    __syncthreads();
#endif

    const int wv   = tid >> 5;     // wave id: column tile
    const int lane = tid & 31;
    const int lm   = lane & 15;    // row/col within tile
    const int lh   = lane >> 4;    // K-half selector per ISA layout
    const int n0   = wv * 16;

    v8f acc = {};
    #pragma unroll
    for (int kk = 0; kk < 32; ++kk) {
        int k = kk * 4 + lh * 2;
        // A fragment: 16x4 f32, lane(lm,lh) holds X[m0+lm][k], X[m0+lm][k+1]
        v2f a = *(const v2f*)(Xs + lm * XS_STRIDE + k);
        // B fragment: 4x16 f32, lane(lm,lh) holds W[k][n0+lm], W[k+1][n0+lm]
        v2f b;
        b.x = Ws[(k    ) * WS_STRIDE + n0 + lm];
        b.y = Ws[(k + 1) * WS_STRIDE + n0 + lm];
        acc = __builtin_amdgcn_wmma_f32_16x16x4_f32(
                  false, a, false, b, (short)0, acc, false, false);
    }

    // C/D layout: VGPR v, lanes 0-15 -> M = v, lanes 16-31 -> M = v+8; N = lm
    #pragma unroll
    for (int v = 0; v < 8; ++v) {
        int m = m0 + v + 8 * lh;
        H[(size_t)m * FDIM + n0 + lm] = acc[v];
    }
}

// ---------------- aggregation ----------------

// agg[i] = h[i] * dis[i]^2  (self-loop term; fully initializes agg buffer)
__global__ void selfloop_kernel(const float* __restrict__ h,
                                const float* __restrict__ dis,
                                float* __restrict__ agg, int n) {
    int t = blockIdx.x * blockDim.x + threadIdx.x;
    if (t >= n * 32) return;
    int node = t >> 5;
    float d = dis[node];
    float s = d * d;
    float4 v = ((const float4*)h)[t];
    float4 r;
    r.x = v.x * s; r.y = v.y * s; r.z = v.z * s; r.w = v.w * s;
    ((float4*)agg)[t] = r;
}

// one wave per edge: agg[dst] += h[src] * dis[src]*dis[dst]
__global__ void edge_kernel(const float* __restrict__ h,
                            const long long* __restrict__ src,
                            const long long* __restrict__ dst,
                            const float* __restrict__ dis,
                            float* __restrict__ agg, int e) {
    int t  = blockIdx.x * blockDim.x + threadIdx.x;
    int ed = t >> 5;
    if (ed >= e) return;
    int lane = t & 31;
    int s = (int)src[ed];
    int d = (int)dst[ed];
    float nrm = dis[s] * dis[d];
    float4 v = ((const float4*)h)[s * 32 + lane];
    float* a = agg + (size_t)d * FDIM + lane * 4;
    unsafeAtomicAdd(a + 0, v.x * nrm);
    unsafeAtomicAdd(a + 1, v.y * nrm);
    unsafeAtomicAdd(a + 2, v.z * nrm);
    unsafeAtomicAdd(a + 3, v.w * nrm);
}

// out = relu(in + bias)
__global__ void bias_relu_kernel(const float* __restrict__ in,
                                 const float* __restrict__ bias,
                                 float* __restrict__ out, int n) {
    int t = blockIdx.x * blockDim.x + threadIdx.x;
    if (t >= n * 32) return;
    int c = t & 31;
    float4 v = ((const float4*)in)[t];
    float4 b = ((const float4*)bias)[c];
    float4 r;
    r.x = fmaxf(v.x + b.x, 0.0f);
    r.y = fmaxf(v.y + b.y, 0.0f);
    r.z = fmaxf(v.z + b.z, 0.0f);
    r.w = fmaxf(v.w + b.w, 0.0f);
    ((float4*)out)[t] = r;
}

// ---------------- launch ----------------

extern "C" void kernel_launch(void* const* d_in, const int* in_sizes, int n_in,
                              void* d_out, int out_size, void* d_ws, size_t ws_size,
                              hipStream_t stream) {
    const float*     x   = (const float*)d_in[0];
    const long long* ei  = (const long long*)d_in[1];   // int64 [2, E]
    const float*     W1  = (const float*)d_in[2];
    const float*     b1  = (const float*)d_in[3];
    const float*     W2  = (const float*)d_in[4];
    const float*     b2  = (const float*)d_in[5];
    const long long* src = ei;
    const long long* dst = ei + NEDGES;

    float* B0  = (float*)d_ws;                          // N x 128
    float* B1  = B0 + (size_t)NNODES * FDIM;            // N x 128
    float* dis = B1 + (size_t)NNODES * FDIM;            // N
    float* out = (float*)d_out;

    const int nBlkN  = (NNODES + 255) / 256;
    const int nBlkE  = (NEDGES + 255) / 256;
    const int nBlkNF = (NNODES * 32) / 256;   // 6250 exact
    const int nBlkEF = (NEDGES * 32) / 256;   // 100000 exact
    const int nBlkG  = NNODES / 16;           // 3125 exact (EXEC all-ones for WMMA)

    // symmetric-normalization coefficients (shared by both layers)
    deg_init_kernel<<<nBlkN, 256, 0, stream>>>(dis, NNODES);
    deg_edge_kernel<<<nBlkE, 256, 0, stream>>>(dst, dis, NEDGES);
    rsqrt_kernel  <<<nBlkN, 256, 0, stream>>>(dis, NNODES);

    // ----- layer 1 -----
    gemm_wmma_kernel<<<nBlkG, 256, 0, stream>>>(x, W1, B0);        // h1 = x @ W1
    selfloop_kernel <<<nBlkNF, 256, 0, stream>>>(B0, dis, B1, NNODES);
    edge_kernel     <<<nBlkEF, 256, 0, stream>>>(B0, src, dst, dis, B1, NEDGES);
    bias_relu_kernel<<<nBlkNF, 256, 0, stream>>>(B1, b1, B0, NNODES);  // act1 -> B0

    // ----- layer 2 -----
    gemm_wmma_kernel<<<nBlkG, 256, 0, stream>>>(B0, W2, B1);       // h2 = act1 @ W2
    selfloop_kernel <<<nBlkNF, 256, 0, stream>>>(B1, dis, out, NNODES);
    edge_kernel     <<<nBlkEF, 256, 0, stream>>>(B1, src, dst, dis, out, NEDGES);
    bias_relu_kernel<<<nBlkNF, 256, 0, stream>>>(out, b2, out, NNODES); // in-place
}